// GaussianSplatting2D_28716151341337
// MI455X (gfx1250) — compile-verified
//
#include <hip/hip_runtime.h>
#include <hip/hip_bf16.h>

typedef __attribute__((ext_vector_type(2))) float v2f;
typedef __attribute__((ext_vector_type(8))) float v8f;

#define N_GAUSS 512
#define N_PIX   262144
// -0.5 * log2(e): fold exp(-q/2) into exp2 of the GEMM output
#define NEG_HALF_LOG2E (-0.72134752044448170367996234050095f)

// ---------------------------------------------------------------------------
// Prep: per-gaussian 8-float record: c0..c5 (scaled, log2(alpha) folded into
// c5), slots 6,7 = 0 (K-padding -> exact rank-6 result from the K=8 WMMA).
// ---------------------------------------------------------------------------
__global__ __launch_bounds__(512) void gs2d_prep(
    const float* __restrict__ alphas,   // (512,1)
    const float* __restrict__ means,    // (512,2)
    const float* __restrict__ rots,     // (512,1)
    const float* __restrict__ scales,   // (512,2)
    float* __restrict__ coef)           // (512,8) out
{
  int n = threadIdx.x;
  if (n >= N_GAUSS) return;

  float rot = rots[n];
  float cth = cosf(rot), sth = sinf(rot);
  float s0 = scales[2*n], s1 = scales[2*n+1];
  // RS = [[s0 c, -s1 s],[s0 s, s1 c]]
  float r00 = s0*cth, r01 = -s1*sth, r10 = s0*sth, r11 = s1*cth;
  // M = RS * RS^T (symmetric)
  float A = r00*r00 + r01*r01;
  float B = r00*r10 + r01*r11;
  float D = r10*r10 + r11*r11;
  // cov = inv(M)
  float det = A*D - B*B;
  float inv = 1.0f / det;
  float a =  D*inv;
  float b = -B*inv;
  float c =  A*inv;

  float mx = means[2*n], my = means[2*n+1];
  float s = NEG_HALF_LOG2E;

  float c0 = s * a;                       // * px^2
  float c1 = s * c;                       // * py^2
  float c2 = s * (2.0f * b);              // * px*py
  float c3 = s * (-2.0f * (a*mx + b*my)); // * px
  float c4 = s * (-2.0f * (b*mx + c*my)); // * py
  float c5 = s * (a*mx*mx + 2.0f*b*mx*my + c*my*my)
           + __builtin_amdgcn_logf(alphas[n]); // v_log_f32 = log2(alpha)

  float* r = coef + n*8;
  r[0] = c0; r[1] = c1; r[2] = c2; r[3] = c3;
  r[4] = c4; r[5] = c5; r[6] = 0.0f; r[7] = 0.0f;
}

// ---------------------------------------------------------------------------
// Main: each wave32 computes a 16(pixel)x512(gaussian) strip.
// Per 16x16 tile: acc = A0*B0 + A1*B1 via two V_WMMA_F32_16X16X4_F32,
// then sum += exp2(acc).  Software-pipelined one tile deep so the 8
// v_exp_f32 (TRANS) ops overlap the next tile's WMMA + DS latency.
// ---------------------------------------------------------------------------
__global__ __launch_bounds__(256) void gs2d_main(
    const float* __restrict__ x,     // (262144,2)
    const float* __restrict__ coef,  // (512,8)
    float* __restrict__ out)         // (262144,)
{
  __shared__ float lds[N_GAUSS * 8]; // 16 KB coefficient table

  {
    const float4* src = (const float4*)coef;
    float4*       dst = (float4*)lds;
    #pragma unroll
    for (int i = 0; i < (N_GAUSS * 8) / 4 / 256; ++i)
      dst[threadIdx.x + i * 256] = src[threadIdx.x + i * 256];
  }
  __syncthreads();

  const int lane = threadIdx.x & 31;
  const int wave = threadIdx.x >> 5;
  const int m0   = (blockIdx.x * 8 + wave) * 16;   // 16 pixels per wave
  const int mp   = m0 + (lane & 15);
  const bool hi  = lane >= 16;

  const float px = x[2*mp];
  const float py = x[2*mp + 1];

  // A-matrix (16x4 f32): lanes 0-15 hold K={0,1}, lanes 16-31 hold K={2,3}.
  // Features: f = [px^2, py^2, px*py, px, py, 1, 0, 0]
  v2f a0, a1;
  if (!hi) { a0.x = px*px;  a0.y = py*py; a1.x = py;   a1.y = 1.0f; }
  else     { a0.x = px*py;  a0.y = px;    a1.x = 0.0f; a1.y = 0.0f; }

  const int kb = hi ? 2 : 0;
  const v2f* lds2 = (const v2f*)lds;
  const int nbase = ((lane & 15) * 8 + kb) >> 1;   // v2f index of this lane's col

  // Packed partial sums (lower add count via v_pk_add_f32)
  v2f psum[4];
  #pragma unroll
  for (int r = 0; r < 4; ++r) psum[r] = (v2f){0.0f, 0.0f};

  // --- tile 0 (prologue) ---
  v8f acc;
  {
    v2f b0 = lds2[nbase];
    v2f b1 = lds2[nbase + 2];
    v8f z = {};
    z   = __builtin_amdgcn_wmma_f32_16x16x4_f32(false, a1, false, b1, (short)0, z,   false, false);
    acc = __builtin_amdgcn_wmma_f32_16x16x4_f32(false, a0, false, b0, (short)0, z,   false, false);
  }

  for (int t = 1; t < N_GAUSS / 16; ++t) {
    // Issue next tile's B-loads + WMMAs first (fills exp latency).
    const int nb = nbase + t * (16 * 8 / 2);
    v2f b0 = lds2[nb];
    v2f b1 = lds2[nb + 2];
    v8f z = {};
    z = __builtin_amdgcn_wmma_f32_16x16x4_f32(false, a1, false, b1, (short)0, z, false, false);
    v8f nacc =
        __builtin_amdgcn_wmma_f32_16x16x4_f32(false, a0, false, b0, (short)0, z, false, false);

    // 8 independent v_exp_f32 into a staging buffer, then packed adds —
    // keeps TRANS results >= 8 instructions away from their consumers.
    float e[8];
    #pragma unroll
    for (int r = 0; r < 8; ++r) e[r] = __builtin_amdgcn_exp2f(acc[r]);
    #pragma unroll
    for (int r = 0; r < 4; ++r) {
      v2f ev = {e[2*r], e[2*r+1]};
      psum[r] += ev;
    }

    acc = nacc;
  }

  // --- epilogue: consume last tile ---
  {
    float e[8];
    #pragma unroll
    for (int r = 0; r < 8; ++r) e[r] = __builtin_amdgcn_exp2f(acc[r]);
    #pragma unroll
    for (int r = 0; r < 4; ++r) {
      v2f ev = {e[2*r], e[2*r+1]};
      psum[r] += ev;
    }
  }

  float sum[8];
  #pragma unroll
  for (int r = 0; r < 4; ++r) { sum[2*r] = psum[r].x; sum[2*r+1] = psum[r].y; }

  // Reduce over the 16 gaussian-column residues held across lanes of each half
  // (wave32: xor masks 1,2,4,8 stay inside each group of 16 lanes).
  #pragma unroll
  for (int off = 1; off < 16; off <<= 1) {
    #pragma unroll
    for (int r = 0; r < 8; ++r)
      sum[r] += __shfl_xor(sum[r], off, 32);
  }

  // Lanes 0 and 16 now hold the full sums for pixel rows m0+0..7 / m0+8..15.
  if ((lane & 15) == 0) {
    float4* o = (float4*)(out + m0 + (hi ? 8 : 0));
    float4 lo4 = {sum[0], sum[1], sum[2], sum[3]};
    float4 hi4 = {sum[4], sum[5], sum[6], sum[7]};
    o[0] = lo4;
    o[1] = hi4;
  }
}

// ---------------------------------------------------------------------------
extern "C" void kernel_launch(void* const* d_in, const int* in_sizes, int n_in,
                              void* d_out, int out_size, void* d_ws, size_t ws_size,
                              hipStream_t stream) {
  const float* x      = (const float*)d_in[0]; // (262144,2)
  const float* alphas = (const float*)d_in[1]; // (512,1)
  const float* means  = (const float*)d_in[2]; // (512,2)
  const float* rots   = (const float*)d_in[3]; // (512,1)
  const float* scales = (const float*)d_in[4]; // (512,2)
  float* out  = (float*)d_out;
  float* coef = (float*)d_ws;                  // 512*8 floats = 16 KB

  gs2d_prep<<<1, 512, 0, stream>>>(alphas, means, rots, scales, coef);
  gs2d_main<<<N_PIX / (16 * 8), 256, 0, stream>>>(x, coef, out);
}